// MultiHeadDecoder_22471268892795
// MI455X (gfx1250) — compile-verified
//
#include <hip/hip_runtime.h>

// Problem constants (from reference)
constexpr int Bq      = 4;
constexpr int Tq      = 12;
constexpr int Vq      = 4;
constexpr int Hq      = 96;
constexpr int Wq      = 144;
constexpr int Dq      = 256;
constexpr int NMID    = 3;
constexpr int NHEADS  = 6;
constexpr int HW      = Hq * Wq;          // 13824 (multiple of 64)
constexpr int NP      = 64;               // pixel-tile width
constexpr int TILES_B = (Tq * HW) / NP;   // 2592 tiles per batch element

typedef __attribute__((ext_vector_type(2))) float v2f;
typedef __attribute__((ext_vector_type(8))) float v8f;

// Fused multi-head 1x1-conv decoder:
//   h = Win[hd] @ x + bin ; 3x (h = Wmid[l][hd] @ h + bmid) ; y = Wout[hd] @ h + bout
// Activation panel [D=256 x NP=64] lives in LDS for the whole chain.
// Mid layers use V_WMMA_F32_16X16X4_F32 (fp32 matrix pipe, wave32).
__global__ __launch_bounds__(256, 2)
void mhd_wmma_kernel(const float* __restrict__ x,
                     const float* __restrict__ Wi,  // [NHEADS, D, V]
                     const float* __restrict__ bi,  // [NHEADS, D]
                     const float* __restrict__ Wm,  // [NMID, NHEADS, D, D]
                     const float* __restrict__ bm,  // [NMID, NHEADS, D]
                     const float* __restrict__ Wo,  // [NHEADS, V, D]
                     const float* __restrict__ bo,  // [NHEADS, V]
                     const int*   __restrict__ hid, // [B]
                     float*       __restrict__ y) {
  __shared__ float Hs[Dq * NP];   // 64 KB activation panel

  const int tIdx = blockIdx.x;            // 0..2591
  const int b    = blockIdx.y;            // 0..3
  const int t    = tIdx / (HW / NP);      // HW/NP = 216
  const int r0   = (tIdx % (HW / NP)) * NP;
  const int hd   = hid[b];

  const int tid  = threadIdx.x;
  const int lane = tid & 31;
  const int wave = tid >> 5;              // 0..7
  const int half = lane >> 4;             // 0: lanes 0-15, 1: lanes 16-31
  const int l16  = lane & 15;

  // base of this tile in x / y: element (v, n) at xb + v*HW + n
  const long xb = ((long)(b * Tq + t) * Vq) * HW + r0;

  // ---------------- input layer: Hs[d,n] = bin[d] + sum_v Win[d,v]*x[v,n]
  {
    const int n  = tid & (NP - 1);        // 0..63
    const int dg = tid >> 6;              // 0..3 -> rows dg*64 .. dg*64+63
    float xv[Vq];
#pragma unroll
    for (int v = 0; v < Vq; ++v) xv[v] = x[xb + (long)v * HW + n];
    const float* wrow = Wi + (hd * Dq + dg * 64) * Vq;
    const float* brow = bi + hd * Dq + dg * 64;
#pragma unroll 4
    for (int dd = 0; dd < 64; ++dd) {
      float acc = brow[dd];
#pragma unroll
      for (int v = 0; v < Vq; ++v) acc += wrow[dd * Vq + v] * xv[v];
      Hs[(dg * 64 + dd) * NP + n] = acc;
    }
  }
  __syncthreads();

  // ---------------- 3 mid layers: Hs <- Wmid[l][hd] (256x256) @ Hs + bias
  for (int l = 0; l < NMID; ++l) {
    const float* Wl = Wm + (long)(l * NHEADS + hd) * Dq * Dq;  // row-major [e][d]
    const float* bl = bm + (l * NHEADS + hd) * Dq;

    // wave owns row-tiles {2w, 2w+1} x all 4 col-tiles -> 8x 16x16 fp32 accs
    v8f acc[2][4] = {};

#pragma unroll 1
    for (int ks = 0; ks < Dq / 4; ++ks) {
      const int k    = ks * 4;
      const int krow = k + 2 * half;      // K rows this lane-half supplies

      // B fragments (4x16 of Hs), shared across both row-tiles:
      // VGPR0 = K (k+2*half), VGPR1 = K+1, N = ct*16 + l16
      v2f bfrag[4];
#pragma unroll
      for (int ct = 0; ct < 4; ++ct) {
        bfrag[ct].x = Hs[krow * NP + ct * 16 + l16];
        bfrag[ct].y = Hs[(krow + 1) * NP + ct * 16 + l16];
      }

#pragma unroll
      for (int i = 0; i < 2; ++i) {
        const int rt = wave * 2 + i;
        // A fragment (16x4 of W): lane row = rt*16+l16, cols k+2*half, +1
        v2f a = *(const v2f*)(Wl + (rt * 16 + l16) * Dq + k + 2 * half);
#pragma unroll
        for (int ct = 0; ct < 4; ++ct) {
          acc[i][ct] = __builtin_amdgcn_wmma_f32_16x16x4_f32(
              false, a, false, bfrag[ct], (short)0, acc[i][ct], false, false);
        }
      }
    }

    __syncthreads();   // all reads of Hs done -> safe to overwrite in place

    // write back + bias; C/D layout: VGPR r -> M = half*8 + r, N = l16
#pragma unroll
    for (int i = 0; i < 2; ++i) {
      const int rt = wave * 2 + i;
#pragma unroll
      for (int ct = 0; ct < 4; ++ct) {
#pragma unroll
        for (int r = 0; r < 8; ++r) {
          const int d = rt * 16 + half * 8 + r;
          Hs[d * NP + ct * 16 + l16] = acc[i][ct][r] + bl[d];
        }
      }
    }
    __syncthreads();
  }

  // ---------------- output layer: y[v,n] = bout[v] + sum_d Wout[v,d]*Hs[d,n]
  {
    const int n = tid & (NP - 1);
    const int v = tid >> 6;               // 0..3
    const float* wrow = Wo + (hd * Vq + v) * Dq;
    float acc = bo[hd * Vq + v];
#pragma unroll 4
    for (int d = 0; d < Dq; ++d) acc += wrow[d] * Hs[d * NP + n];
    y[xb + (long)v * HW + n] = acc;
  }
}

extern "C" void kernel_launch(void* const* d_in, const int* in_sizes, int n_in,
                              void* d_out, int out_size, void* d_ws, size_t ws_size,
                              hipStream_t stream) {
  (void)in_sizes; (void)n_in; (void)out_size; (void)d_ws; (void)ws_size;
  const float* x  = (const float*)d_in[0];
  const float* Wi = (const float*)d_in[1];
  const float* bi = (const float*)d_in[2];
  const float* Wm = (const float*)d_in[3];
  const float* bm = (const float*)d_in[4];
  const float* Wo = (const float*)d_in[5];
  const float* bo = (const float*)d_in[6];
  const int*  hid = (const int*)d_in[7];
  float* y = (float*)d_out;

  dim3 grid(TILES_B, Bq);
  dim3 block(256);
  mhd_wmma_kernel<<<grid, block, 0, stream>>>(x, Wi, bi, Wm, bm, Wo, bo, hid, y);
}